// Model_8976481648805
// MI455X (gfx1250) — compile-verified
//
#include <hip/hip_runtime.h>
#include <math.h>

// ---------------- model constants ----------------
#define NH     256      // HiPPO state dim
#define PPAIRS 256      // B*ENC_IN = 8*32
#define SEQ    768
#define PRED   192
#define MODES  32
#define COMPN  16
#define TWO_PI 6.28318530717958647692f

typedef __attribute__((ext_vector_type(2))) float v2f;
typedef __attribute__((ext_vector_type(8))) float v8f;

// ---------------- workspace layout (float offsets) ----------------
#define WS_MEAN   0                          // 256
#define WS_STD    (WS_MEAN + 256)            // 256
#define WS_FNORM  (WS_STD + 256)             // 256*768
#define WS_AT     (WS_FNORM + 256*768)       // 3 * 256*256   (A^T per scale)
#define WS_K      (WS_AT + 3*65536)          // 3 * 768*256   (K chain)
#define WS_KW     (WS_K + 3*196608)          // 3 * 768*32    (K*w0, complex)
#define WS_Q      (WS_KW + 3*24576)          // 3 * 32*768*16*2 (prefix sums)
#define WS_W2     (WS_Q + 3*786432)          // 3 * 768*1024  (GEMM2 B-matrix)
#define WS_ABUF   (WS_W2 + 3*786432)         // 3 * 256*1024
#define WS_TMAT   (WS_ABUF + 3*262144)       // 3 * 1024*256
#define WS_OUT1   (WS_TMAT + 3*262144)       // 256*768
#define WS_BCAT   (WS_OUT1 + 256*768)        // 768*192
#define WS_XDEC   (WS_BCAT + 768*192)        // 256*192

// =================================================================
// Register-blocked fp32 WMMA GEMM: each wave computes an MTx NT block
// of 16x16 tiles of C = A(MxK) * B(KxN).
// grid = (N/(16*NT), M/(16*MT)), block = 32 (one wave). K % 16 == 0.
// V_WMMA_F32_16X16X4_F32 fragment layout (ISA 7.12.2):
//   A 16x4: lane<16 -> M=lane, {v0,v1}={K0,K1}; lane>=16 -> {K2,K3}
//   B 4x16: lane<16 -> N=lane, {v0,v1}={K0,K1}; lane>=16 -> {K2,K3}
//   C/D:    vgpr j : lane<16 -> M=j, N=lane ; lane>=16 -> M=j+8
// =================================================================
template <int MT, int NT>
__global__ void __launch_bounds__(32)
wmma_gemm_f32(const float* __restrict__ A, int lda,
              const float* __restrict__ B, int ldb,
              float* __restrict__ C, int ldc, int K)
{
    const int tn   = blockIdx.x * (16 * NT);
    const int tm   = blockIdx.y * (16 * MT);
    const int lane = threadIdx.x;
    const int half = lane >> 4;
    const int l16  = lane & 15;

    const float* Ar = A + (size_t)(tm + l16) * lda + 2 * half;
    const float* Bc = B + (size_t)(2 * half) * ldb + tn + l16;

    v8f acc[MT][NT];
#pragma unroll
    for (int mi = 0; mi < MT; ++mi)
#pragma unroll
        for (int ni = 0; ni < NT; ++ni)
            acc[mi][ni] = (v8f){0.f, 0.f, 0.f, 0.f, 0.f, 0.f, 0.f, 0.f};

    for (int kk = 0; kk < K; kk += 16) {
#pragma unroll
        for (int q = 0; q < 4; ++q) {
            const int k0 = kk + 4 * q;
            v2f a[MT], b[NT];
#pragma unroll
            for (int mi = 0; mi < MT; ++mi)
                a[mi] = (v2f){ Ar[(size_t)mi * 16 * lda + k0],
                               Ar[(size_t)mi * 16 * lda + k0 + 1] };
#pragma unroll
            for (int ni = 0; ni < NT; ++ni)
                b[ni] = (v2f){ Bc[(size_t)k0 * ldb + ni * 16],
                               Bc[(size_t)(k0 + 1) * ldb + ni * 16] };
#pragma unroll
            for (int mi = 0; mi < MT; ++mi)
#pragma unroll
                for (int ni = 0; ni < NT; ++ni)
                    acc[mi][ni] = __builtin_amdgcn_wmma_f32_16x16x4_f32(
                        false, a[mi], false, b[ni], (short)0, acc[mi][ni],
                        false, false);
        }
    }

#pragma unroll
    for (int mi = 0; mi < MT; ++mi) {
        float* Cp = C + (size_t)(tm + mi * 16 + 8 * half) * ldc + tn + l16;
#pragma unroll
        for (int ni = 0; ni < NT; ++ni)
#pragma unroll
            for (int j = 0; j < 8; ++j)
                Cp[(size_t)j * ldc + ni * 16] = acc[mi][ni][j];
    }
}

// =================================================================
// RevIN normalize: per pair p=(b,e): mean/std over 768 steps,
// fnorm[p][s] = (x-mean)/std * aw[e] + ab[e]
// =================================================================
__global__ void __launch_bounds__(256)
norm_kernel(const float* __restrict__ x, const float* __restrict__ aw,
            const float* __restrict__ ab, float* __restrict__ meanA,
            float* __restrict__ stdA, float* __restrict__ fnorm)
{
    __shared__ float red[256];
    const int p = blockIdx.x, b = p >> 5, e = p & 31, tid = threadIdx.x;
    const float v0 = x[(size_t)(b * SEQ + tid      ) * 32 + e];
    const float v1 = x[(size_t)(b * SEQ + tid + 256) * 32 + e];
    const float v2 = x[(size_t)(b * SEQ + tid + 512) * 32 + e];
    red[tid] = v0 + v1 + v2;
    __syncthreads();
    for (int st = 128; st > 0; st >>= 1) {
        if (tid < st) red[tid] += red[tid + st];
        __syncthreads();
    }
    const float mean = red[0] * (1.f / 768.f);
    __syncthreads();
    const float d0 = v0 - mean, d1 = v1 - mean, d2 = v2 - mean;
    red[tid] = d0 * d0 + d1 * d1 + d2 * d2;
    __syncthreads();
    for (int st = 128; st > 0; st >>= 1) {
        if (tid < st) red[tid] += red[tid + st];
        __syncthreads();
    }
    const float stdev = sqrtf(red[0] * (1.f / 768.f) + 1e-5f);
    if (tid == 0) { meanA[p] = mean; stdA[p] = stdev; }
    const float w = aw[e], bb = ab[e], rs = 1.f / stdev;
    float* f = fnorm + (size_t)p * SEQ;
    f[tid]       = d0 * rs * w + bb;
    f[tid + 256] = d1 * rs * w + bb;
    f[tid + 512] = d2 * rs * w + bb;
}

// transpose A -> A^T for the 3 scales
__global__ void transpose_kernel(const float* A0, const float* A1,
                                 const float* A2, float* Atb)
{
    int idx = blockIdx.x * blockDim.x + threadIdx.x;   // 3*65536
    if (idx >= 3 * 65536) return;
    const int s = idx / 65536, r = idx % 65536, m = r / 256, n = r % 256;
    const float* A = (s == 0) ? A0 : (s == 1) ? A1 : A2;
    Atb[(size_t)s * 65536 + m * 256 + n] = A[n * 256 + m];
}

// serial kernel chain: K[0]=Bvec, K[tau] = K[tau-1] @ A^T
__global__ void __launch_bounds__(256)
kchain_kernel(const float* __restrict__ Atb, const float* B0,
              const float* B1, const float* B2, float* __restrict__ Kb)
{
    __shared__ float cur[256];
    const int s = blockIdx.x, n = threadIdx.x;
    const int L = 192 << s;
    const float* At = Atb + (size_t)s * 65536;
    const float* Bv = (s == 0) ? B0 : (s == 1) ? B1 : B2;
    float* K = Kb + (size_t)s * 196608;
    float c = Bv[n];
    cur[n] = c;
    K[n] = c;
    __syncthreads();
    for (int tau = 1; tau < L; ++tau) {
        float acc = 0.f;
        for (int m = 0; m < 256; ++m) acc += cur[m] * At[m * 256 + n];
        __syncthreads();
        cur[n] = acc;
        K[(size_t)tau * 256 + n] = acc;
        __syncthreads();
    }
}

// prefix DFT sums: Q[k][u][h] = sum_{tau<=u} e^{-i w k tau} KW[tau][h]
__global__ void __launch_bounds__(512)
prefix_kernel(const float* __restrict__ KWb, float* __restrict__ Qb)
{
    const int s = blockIdx.x, j = threadIdx.x;      // j in [0,512)
    const int L = 192 << s;
    const int k = j >> 4, h = j & 15;
    const float* KW = KWb + (size_t)s * 24576;
    float* Q = Qb + (size_t)s * 786432;
    const float w = -TWO_PI * (float)k / (float)L;
    float qr = 0.f, qi = 0.f;
    for (int u = 0; u < L; ++u) {
        float si, co;
        sincosf(w * (float)u, &si, &co);
        const float ar = KW[u * 32 + h * 2], ai = KW[u * 32 + h * 2 + 1];
        qr += ar * co - ai * si;
        qi += ar * si + ai * co;
        Q[((size_t)(k * L + u)) * 32 + h * 2]     = qr;
        Q[((size_t)(k * L + u)) * 32 + h * 2 + 1] = qi;
    }
}

// W2[s][(h*32+k)*2+ri] = e^{-i w k s} * Q[k][L-1-s][h]
__global__ void w2_kernel(const float* __restrict__ Qb,
                          float* __restrict__ W2b, int scale)
{
    const int L = 192 << scale;
    int idx = blockIdx.x * blockDim.x + threadIdx.x;
    if (idx >= L * 512) return;
    const int s = idx >> 9, j = idx & 511, h = j >> 5, k = j & 31;
    const float* Q = Qb + (size_t)scale * 786432;
    float* W2 = W2b + (size_t)scale * 786432;
    float si, co;
    sincosf(-TWO_PI * (float)k * (float)s / (float)L, &si, &co);
    const int u = L - 1 - s;
    const float qr = Q[((size_t)(k * L + u)) * 32 + h * 2];
    const float qi = Q[((size_t)(k * L + u)) * 32 + h * 2 + 1];
    W2[(size_t)s * 1024 + (h * 32 + k) * 2]     = qr * co - qi * si;
    W2[(size_t)s * 1024 + (h * 32 + k) * 2 + 1] = qr * si + qi * co;
}

// Tmat rows [(h*32+k)*2+{0,1}] = [Re, -Im] of c_k * sum_kp w1[h,kp,k]*w2[kp,n]
__global__ void tmat_kernel(const float* __restrict__ w1,
                            const float* __restrict__ w2,
                            float* __restrict__ Tm, int L)
{
    int idx = blockIdx.x * blockDim.x + threadIdx.x;   // 16*32*256
    if (idx >= 131072) return;
    const int n = idx & 255, k = (idx >> 8) & 31, h = idx >> 13;
    float ar = 0.f, ai = 0.f;
    for (int kp = 0; kp < 16; ++kp) {
        const float wr = w1[(((h * 16 + kp) * 32 + k) << 1)];
        const float wi = w1[(((h * 16 + kp) * 32 + k) << 1) + 1];
        const float vr = w2[((kp * 256 + n) << 1)];
        const float vi = w2[((kp * 256 + n) << 1) + 1];
        ar += wr * vr - wi * vi;
        ai += wr * vi + wi * vr;
    }
    const float sc = ((k == 0) ? 1.f : 2.f) / (float)L;
    float si, co;
    sincosf(TWO_PI * (float)k * 191.f / (float)L, &si, &co);
    Tm[(size_t)((h * 32 + k) * 2)     * 256 + n] =  sc * (ar * co - ai * si);
    Tm[(size_t)((h * 32 + k) * 2 + 1) * 256 + n] = -sc * (ar * si + ai * co);
}

// Bcat[i*256+n][t] = mlp_w[i] * eval_i[L_i-192+t][n]
__global__ void bcat_kernel(const float* e0, const float* e1, const float* e2,
                            const float* __restrict__ mw, float* __restrict__ Bc)
{
    int idx = blockIdx.x * blockDim.x + threadIdx.x;   // 768*192
    if (idx >= 768 * 192) return;
    const int t = idx % 192, row = idx / 192, sc = row >> 8, n = row & 255;
    const float* ev = (sc == 0) ? e0 : (sc == 1) ? e1 : e2;
    const int L = 192 << sc;
    Bc[idx] = mw[sc] * ev[(size_t)(L - 192 + t) * 256 + n];
}

// denormalize to output (B, 192, 32)
__global__ void final_kernel(const float* __restrict__ xdec,
                             const float* __restrict__ mb,
                             const float* __restrict__ aw,
                             const float* __restrict__ ab,
                             const float* __restrict__ meanA,
                             const float* __restrict__ stdA,
                             float* __restrict__ out)
{
    int idx = blockIdx.x * blockDim.x + threadIdx.x;   // 8*192*32
    if (idx >= 49152) return;
    const int e = idx & 31, t = (idx >> 5) % 192, b = idx / (192 * 32);
    const int p = b * 32 + e;
    float v = xdec[(size_t)p * 192 + t] + mb[0];
    v = (v - ab[e]) / (aw[e] + 1e-10f);
    out[idx] = v * stdA[p] + meanA[p];
}

// =================================================================
extern "C" void kernel_launch(void* const* d_in, const int* in_sizes, int n_in,
                              void* d_out, int out_size, void* d_ws, size_t ws_size,
                              hipStream_t stream)
{
    const float* x_enc = (const float*)d_in[0];
    const float* aw    = (const float*)d_in[4];
    const float* ab    = (const float*)d_in[5];
    const float* A[3]  = {(const float*)d_in[6],  (const float*)d_in[12], (const float*)d_in[18]};
    const float* Bv[3] = {(const float*)d_in[7],  (const float*)d_in[13], (const float*)d_in[19]};
    const float* ev[3] = {(const float*)d_in[8],  (const float*)d_in[14], (const float*)d_in[20]};
    const float* w0[3] = {(const float*)d_in[9],  (const float*)d_in[15], (const float*)d_in[21]};
    const float* w1[3] = {(const float*)d_in[10], (const float*)d_in[16], (const float*)d_in[22]};
    const float* w2[3] = {(const float*)d_in[11], (const float*)d_in[17], (const float*)d_in[23]};
    const float* mlp_w = (const float*)d_in[24];
    const float* mlp_b = (const float*)d_in[25];
    float* out = (float*)d_out;
    float* ws  = (float*)d_ws;

    float* MEAN  = ws + WS_MEAN;
    float* STD   = ws + WS_STD;
    float* FNORM = ws + WS_FNORM;
    float* AT    = ws + WS_AT;
    float* KCH   = ws + WS_K;
    float* KW    = ws + WS_KW;
    float* Q     = ws + WS_Q;
    float* W2B   = ws + WS_W2;
    float* ABUF  = ws + WS_ABUF;
    float* TMAT  = ws + WS_TMAT;
    float* OUT1  = ws + WS_OUT1;
    float* BCAT  = ws + WS_BCAT;
    float* XDEC  = ws + WS_XDEC;

    // 1) RevIN normalize
    norm_kernel<<<256, 256, 0, stream>>>(x_enc, aw, ab, MEAN, STD, FNORM);

    // 2) A^T transposes + serial K chains
    transpose_kernel<<<(3 * 65536 + 255) / 256, 256, 0, stream>>>(A[0], A[1], A[2], AT);
    kchain_kernel<<<3, 256, 0, stream>>>(AT, Bv[0], Bv[1], Bv[2], KCH);

    // 3) GEMM1 per scale: KW (Lx32) = K (Lx256) * w0 (256x32)  [w0 is (n,h,ri)]
    for (int s = 0; s < 3; ++s) {
        const int L = 192 << s;
        wmma_gemm_f32<2, 2><<<dim3(1, L / 32), 32, 0, stream>>>(
            KCH + (size_t)s * 196608, 256, w0[s], 32, KW + (size_t)s * 24576, 32, 256);
    }

    // 4) prefix DFT sums over tau (all scales)
    prefix_kernel<<<3, 512, 0, stream>>>(KW, Q);

    for (int s = 0; s < 3; ++s) {
        const int L = 192 << s;
        // 5) W2 (Lx1024)
        w2_kernel<<<(L * 512 + 255) / 256, 256, 0, stream>>>(Q, W2B, s);
        // 6) Tmat (1024x256) folds w1, w2, and single-point irfft phase
        tmat_kernel<<<512, 256, 0, stream>>>(w1[s], w2[s], TMAT + (size_t)s * 262144, L);
        // 7) GEMM2: a (256x1024) = fnorm_tail (256xL) * W2 (Lx1024)
        wmma_gemm_f32<2, 4><<<dim3(16, 8), 32, 0, stream>>>(
            FNORM + (SEQ - L), SEQ, W2B + (size_t)s * 786432, 1024,
            ABUF + (size_t)s * 262144, 1024, L);
        // 8) GEMM3: out1 block (256x256) = a (256x1024) * Tmat (1024x256)
        wmma_gemm_f32<2, 4><<<dim3(4, 8), 32, 0, stream>>>(
            ABUF + (size_t)s * 262144, 1024, TMAT + (size_t)s * 262144, 256,
            OUT1 + s * 256, 768, 1024);
    }

    // 9) Legendre read-out with mlp fold: xdec (256x192) = OUT1 (256x768) * BCAT (768x192)
    bcat_kernel<<<(768 * 192 + 255) / 256, 256, 0, stream>>>(ev[0], ev[1], ev[2], mlp_w, BCAT);
    wmma_gemm_f32<2, 4><<<dim3(3, 8), 32, 0, stream>>>(OUT1, 768, BCAT, 192, XDEC, 192, 768);

    // 10) denormalize
    final_kernel<<<(49152 + 255) / 256, 256, 0, stream>>>(XDEC, mlp_b, aw, ab, MEAN, STD, out);

    (void)in_sizes; (void)n_in; (void)out_size; (void)ws_size;
}